// WaveformDifference_80891414053121
// MI455X (gfx1250) — compile-verified
//
#include <hip/hip_runtime.h>

// Problem shape from the reference setup (B,T,D) = (8,1024,8), fp32.
static constexpr int BATCH = 8;
static constexpr int T = 1024;
static constexpr int DIM = 8;
static constexpr int TILE = 32;
static constexpr int NT = T / TILE;   // 32 tiles per side
static constexpr int NWAVE = 32;      // waves per 1024-thread block
static constexpr float EPSV = 1e-6f;
static constexpr float BIGV = 1e30f;

typedef __attribute__((ext_vector_type(2))) float v2f;
typedef __attribute__((ext_vector_type(8))) float v8f;

__device__ __forceinline__ float shfl_up1(float x) {
    // dst[lane] = src[lane-1]; lane 0 result is unused (boundary path taken).
    const int li = (int)(threadIdx.x & 31);
    const int idx = (li - 1) << 2;
    return __int_as_float(__builtin_amdgcn_ds_bpermute(idx, __float_as_int(x)));
}

__global__ __launch_bounds__(1024, 1)
void dtw_wmma_kernel(const float* __restrict__ obs_all,
                     const float* __restrict__ mod_all,
                     float* __restrict__ ws) {
    __shared__ float distT[NWAVE][TILE][TILE];   // 128 KB: per-wave dist tile
    __shared__ float no2T[T];                    // ||o_i||^2 + 2*eps*sum(o_i)
    __shared__ float nmT[T];                     // ||m_j||^2 - 2*eps*sum(m_j)
    __shared__ float topRow[T];                  // bottom row of last tile per column
    __shared__ float leftCol[T];                 // right col of last tile per row
    __shared__ float cornerBuf[2][NT];           // parity-buffered tile corners

    const int t = (int)threadIdx.x;
    const int w = t >> 5;                        // wave id == tile row ti
    const int lane = t & 31;
    const int halfSel = (lane >> 4) & 1;         // WMMA half-wave select
    const int lm = lane & 15;
    const int b = (int)blockIdx.x;

    const float* ob = obs_all + (size_t)b * T * DIM;
    const float* mo = mod_all + (size_t)b * T * DIM;

    // ---- Phase 0: per-row norm terms + boundary init ----
    {
        float s2o = 0.f, s1o = 0.f, s2m = 0.f, s1m = 0.f;
#pragma unroll
        for (int d = 0; d < DIM; ++d) {
            const float vo = ob[t * DIM + d];
            const float vm = mo[t * DIM + d];
            s2o += vo * vo; s1o += vo;
            s2m += vm * vm; s1m += vm;
        }
        no2T[t] = s2o + 2.f * EPSV * s1o;
        nmT[t]  = s2m - 2.f * EPSV * s1m;
        topRow[t] = BIGV;
        leftCol[t] = BIGV;
        if (t < 2 * NT) ((float*)cornerBuf)[t] = BIGV;
    }
    __syncthreads();

    const int ti = w;
    const int i0 = ti * TILE;

    // ---- Blocked wavefront over 63 tile-diagonals ----
    for (int d = 0; d < 2 * NT - 1; ++d) {
        const int tj = d - ti;
        const bool valid = (tj >= 0) && (tj < NT);
        const int j0 = tj * TILE;
        float cornerIn = BIGV;

        if (valid) {
            if (ti > 0 && tj > 0) cornerIn = cornerBuf[d & 1][tj - 1];

            // --- 32x32 distance tile via 8x V_WMMA_F32_16X16X4_F32 ---
            const int kb = halfSel * 2;          // K base within 4-chunk
            v8f acc[2][2];
            {
                v8f z = {};
#pragma unroll
                for (int mi = 0; mi < 2; ++mi)
#pragma unroll
                    for (int nj = 0; nj < 2; ++nj) acc[mi][nj] = z;
            }
#pragma unroll
            for (int kc = 0; kc < 2; ++kc) {
                v2f a[2], bf[2];
#pragma unroll
                for (int mi = 0; mi < 2; ++mi) {
                    const float* p = ob + (size_t)(i0 + 16 * mi + lm) * DIM + kc * 4 + kb;
                    a[mi].x = p[0]; a[mi].y = p[1];
                }
#pragma unroll
                for (int nj = 0; nj < 2; ++nj) {
                    const float* p = mo + (size_t)(j0 + 16 * nj + lm) * DIM + kc * 4 + kb;
                    bf[nj].x = p[0]; bf[nj].y = p[1];
                }
#pragma unroll
                for (int mi = 0; mi < 2; ++mi)
#pragma unroll
                    for (int nj = 0; nj < 2; ++nj)
                        acc[mi][nj] = __builtin_amdgcn_wmma_f32_16x16x4_f32(
                            false, a[mi], false, bf[nj], (short)0, acc[mi][nj],
                            false, false);
            }

            // Convert Gram entries to L2 distances, stage tile into LDS.
            const float cEps = (float)DIM * EPSV * EPSV;
#pragma unroll
            for (int mi = 0; mi < 2; ++mi) {
#pragma unroll
                for (int nj = 0; nj < 2; ++nj) {
                    const int N = 16 * nj + lm;
                    const float bn = nmT[j0 + N];
#pragma unroll
                    for (int r = 0; r < 8; ++r) {
                        const int M = 16 * mi + 8 * halfSel + r;
                        float d2 = no2T[i0 + M] + bn - 2.f * acc[mi][nj][r] + cEps;
                        distT[w][M][N] = sqrtf(fmaxf(d2, 0.f));
                    }
                }
            }
        }
        __syncthreads();   // corner reads & dist staging done; boundaries stable

        if (valid) {
            const int li = lane;                 // lane == tile-local row
            float prev = BIGV, prev2 = BIGV;     // diagonals kk-1, kk-2
            for (int kk = 0; kk < 2 * TILE - 1; ++kk) {
                const int lj = kk - li;
                const bool inR = (lj >= 0) && (lj < TILE);
                const int ljc = inR ? lj : 0;
                const float dd = distT[w][li][ljc];

                const float pUp  = shfl_up1(prev);   // dtw[li-1, lj]
                const float p2Up = shfl_up1(prev2);  // dtw[li-1, lj-1]

                const float up   = (li == 0) ? topRow[j0 + ljc] : pUp;
                const float left = (lj == 0) ? leftCol[i0 + li] : prev;
                float dg;
                if (li == 0) dg = (lj == 0) ? cornerIn : topRow[j0 + ljc - 1];
                else         dg = (lj == 0) ? leftCol[i0 + li - 1] : p2Up;

                float best = fminf(fminf(up, left), dg);
                if (i0 == 0 && j0 == 0 && li == 0 && lj == 0) best = 0.f; // dtw[0,0]=dist[0,0]
                const float cur = inR ? (dd + best) : BIGV;

                if (inR) {
                    if (li == TILE - 1) topRow[j0 + lj] = cur;       // export bottom row
                    if (lj == TILE - 1) leftCol[i0 + li] = cur;      // export right col
                    if (li == TILE - 1 && lj == TILE - 1)
                        cornerBuf[d & 1][tj] = cur;                  // export corner
                }
                prev2 = prev;
                prev = cur;
            }
        }
        __syncthreads();   // publish boundaries for diagonal d+1
    }

    if (t == 0) ws[b] = cornerBuf[(2 * NT - 2) & 1][NT - 1];  // dtw[T-1,T-1]
}

__global__ void dtw_finalize_kernel(const float* __restrict__ ws,
                                    float* __restrict__ out) {
    if (threadIdx.x == 0) {
        float s = 0.f;
#pragma unroll
        for (int i = 0; i < BATCH; ++i) s += ws[i];
        out[0] = s * (1.0f / (float)BATCH);
    }
}

extern "C" void kernel_launch(void* const* d_in, const int* in_sizes, int n_in,
                              void* d_out, int out_size, void* d_ws, size_t ws_size,
                              hipStream_t stream) {
    (void)in_sizes; (void)n_in; (void)out_size; (void)ws_size;
    const float* obs = (const float*)d_in[0];
    const float* mod = (const float*)d_in[1];
    float* ws = (float*)d_ws;

    dtw_wmma_kernel<<<BATCH, 1024, 0, stream>>>(obs, mod, ws);
    dtw_finalize_kernel<<<1, 32, 0, stream>>>(ws, (float*)d_out);
}